// CTC_28819230556189
// MI455X (gfx1250) — compile-verified
//
#include <hip/hip_runtime.h>
#include <cstdint>

// Problem constants (from reference setup_inputs)
#define Bsz   32
#define Tlen  500
#define Edim  1024
#define Vocab 5000
#define Llen  100
#define Sext  201      // 2*L+1
#define Spad  208      // padded to 13 * 16
#define NEGV  (-1.0e30f)

// GEMM tiling
#define APITCH 1028    // 1024 + 4 pad -> conflict-free column reads
#define WPITCH 68      // 64 + 4 pad
#define NSPLIT 4
#define NG_TOTAL 79    // ceil(5000/64) 64-column groups
#define NG_PER 20      // ceil(79/4)

typedef __attribute__((ext_vector_type(2))) float v2f;
typedef __attribute__((ext_vector_type(8))) float v8f;

__device__ __forceinline__ float rmax16(float x) {
  x = fmaxf(x, __shfl_xor(x, 1, 16));
  x = fmaxf(x, __shfl_xor(x, 2, 16));
  x = fmaxf(x, __shfl_xor(x, 4, 16));
  x = fmaxf(x, __shfl_xor(x, 8, 16));
  return x;
}
__device__ __forceinline__ float rsum16(float x) {
  x += __shfl_xor(x, 1, 16);
  x += __shfl_xor(x, 2, 16);
  x += __shfl_xor(x, 4, 16);
  x += __shfl_xor(x, 8, 16);
  return x;
}

__device__ __forceinline__ void async_wait0() {
#if __has_builtin(__builtin_amdgcn_s_wait_asynccnt)
  __builtin_amdgcn_s_wait_asynccnt(0);
#else
  asm volatile("s_wait_asynccnt 0x0" ::: "memory");
#endif
}

// Async DMA of a 64-row x 64-float W chunk straight into LDS (no VGPR data).
// GVS mode: SGPR 64-bit base + per-lane 32-bit byte offset (W is 20.5MB, fits).
// Rows >= Vocab are clamped to the last row; those columns are masked later.
__device__ __forceinline__ void loadW_async(unsigned long long Wbase,
                                            unsigned ldsBase, int N0, int kOff,
                                            int tid) {
#pragma unroll
  for (int it = 0; it < 8; ++it) {
    int idx = tid + it * 128;          // 0..1023 (64 rows x 16 b128 slots)
    int i = idx >> 4;                  // chunk row (output column N0+i)
    int c = idx & 15;                  // 16-byte slot within row
    int v = N0 + i; if (v > Vocab - 1) v = Vocab - 1;
    unsigned goff  = ((unsigned)v << 12) + (unsigned)((kOff + (c << 2)) << 2);
    unsigned laddr = ldsBase + (unsigned)(i * (WPITCH * 4) + (c << 4));
    asm volatile("global_load_async_to_lds_b128 %0, %1, %2 offset:0"
                 :: "v"(laddr), "v"(goff), "s"(Wbase) : "memory");
  }
}

// ---------------------------------------------------------------------------
// Kernel 1: fp32 WMMA GEMM over all of V, producing per-row online softmax
// stats (max m, sum s) for a V-slice. grid = (250 row-tiles, NSPLIT), 128 thr.
// A tile (64 rows x 1024) is LDS-resident; W streamed via async DMA in
// double-buffered 64x64 chunks. Each wave: 16 rows x 64 cols (4 accumulators).
// ---------------------------------------------------------------------------
__global__ void __launch_bounds__(128)
gemm_rowstats_kernel(const float* __restrict__ A, const float* __restrict__ W,
                     float* __restrict__ part) {
  extern __shared__ float sm1[];
  float* As  = sm1;                  // 64 x APITCH
  float* Wsh = sm1 + 64 * APITCH;    // 2 x 64 x WPITCH

  const int tid  = threadIdx.x;
  const int w    = tid >> 5;         // wave id (0..3) -> row group
  const int lane = tid & 31;
  const int half = lane >> 4;
  const int li   = lane & 15;
  const int wg   = blockIdx.x;       // row tile (64 rows)
  const int j    = blockIdx.y;       // V split

  const unsigned wshBase = (unsigned)(uintptr_t)Wsh;
  const unsigned long long Wbase = (unsigned long long)(uintptr_t)W;

  // Stage A tile: 64 rows x 1024 floats (coalesced float4)
  for (int idx = tid; idx < 64 * 256; idx += 128) {
    int r  = idx >> 8;
    int c4 = (idx & 255) << 2;
    float4 v = *(const float4*)(A + (size_t)(wg * 64 + r) * Edim + c4);
    *(float4*)(As + r * APITCH + c4) = v;
  }

  float m[8], ssum[8];
#pragma unroll
  for (int i = 0; i < 8; ++i) { m[i] = NEGV; ssum[i] = 0.0f; }

  const int ng0 = j * NG_PER;
  const int ng1 = min(NG_TOTAL, ng0 + NG_PER);

  // First W chunk in flight; barrier also covers the A-tile ds_stores.
  loadW_async(Wbase, wshBase, ng0 * 64, 0, tid);
  async_wait0();
  __syncthreads();

  int cur = 0;
  for (int ng = ng0; ng < ng1; ++ng) {
    const int N0 = ng * 64;
    v8f acc[4];
#pragma unroll
    for (int c = 0; c < 4; ++c)
      acc[c] = (v8f){0.f,0.f,0.f,0.f,0.f,0.f,0.f,0.f};

    for (int kb = 0; kb < 16; ++kb) {
      // Continuous prefetch: next chunk of this group, or chunk 0 of next group
      int pn = ng, pk = kb + 1;
      if (pk == 16) { pn = ng + 1; pk = 0; }
      if (pn < ng1)
        loadW_async(Wbase, wshBase + (unsigned)((cur ^ 1) * 64 * WPITCH * 4),
                    pn * 64, pk * 64, tid);

      // 64 WMMAs (K=64, 4 column tiles) on current buffer
      const float* Ap = As + (w * 16 + li) * APITCH + kb * 64 + 2 * half;
      const float* Bp = Wsh + cur * (64 * WPITCH) + li * WPITCH + 2 * half;
#pragma unroll
      for (int kk = 0; kk < 16; ++kk) {
        v2f a = *(const v2f*)(Ap + kk * 4);
#pragma unroll
        for (int c = 0; c < 4; ++c) {
          v2f bq = *(const v2f*)(Bp + c * (16 * WPITCH) + kk * 4);
          acc[c] = __builtin_amdgcn_wmma_f32_16x16x4_f32(false, a, false, bq,
                                                         (short)0, acc[c],
                                                         false, false);
        }
      }
      async_wait0();
      __syncthreads();
      cur ^= 1;
    }

    // Online (flash) softmax update over the four 16-col tiles.
#pragma unroll
    for (int c = 0; c < 4; ++c) {
      const bool valid = (N0 + c * 16 + li) < Vocab;
#pragma unroll
      for (int i = 0; i < 8; ++i) {
        float x  = valid ? acc[c][i] : NEGV;
        float tm = rmax16(x);
        float mn = fmaxf(m[i], tm);
        float ts = rsum16(expf(x - mn));
        ssum[i] = ssum[i] * expf(m[i] - mn) + ts;
        m[i] = mn;
      }
    }
  }

  if (li == 0) {
    const int R0 = wg * 64 + w * 16 + (half ? 8 : 0);
#pragma unroll
    for (int i = 0; i < 8; ++i) {
      part[((size_t)(R0 + i) * NSPLIT + j) * 2 + 0] = m[i];
      part[((size_t)(R0 + i) * NSPLIT + j) * 2 + 1] = ssum[i];
    }
  }
}

// ---------------------------------------------------------------------------
// Kernel 1m: merge V-split partial (m, s) -> rowlse[r] = m* + log(sum)
// ---------------------------------------------------------------------------
__global__ void rowstats_merge_kernel(const float* __restrict__ part,
                                      float* __restrict__ rowlse) {
  int r = blockIdx.x * blockDim.x + threadIdx.x;
  if (r >= Bsz * Tlen) return;
  float mbest = NEGV;
  for (int j = 0; j < NSPLIT; ++j)
    mbest = fmaxf(mbest, part[((size_t)r * NSPLIT + j) * 2]);
  float z = 0.f;
  for (int j = 0; j < NSPLIT; ++j) {
    float mj = part[((size_t)r * NSPLIT + j) * 2];
    float sj = part[((size_t)r * NSPLIT + j) * 2 + 1];
    z += sj * expf(mj - mbest);
  }
  rowlse[r] = mbest + logf(z);
}

// ---------------------------------------------------------------------------
// Kernel 2: gathered-label WMMA GEMM: lp[b,t,s] = <hpad[b,t], W[ext[b,s]]>
//           + bias[ext] - rowlse[b,t].  grid = (32 t-tiles, 32 b), 128 thr.
// ---------------------------------------------------------------------------
__global__ void __launch_bounds__(128)
gemm_labels_kernel(const float* __restrict__ A, const float* __restrict__ W,
                   const float* __restrict__ bias, const int* __restrict__ ys,
                   const float* __restrict__ rowlse, float* __restrict__ lp) {
  extern __shared__ float sm2[];
  float* As = sm2;                   // 16 x WPITCH
  float* Ws = sm2 + 16 * WPITCH;     // Spad x WPITCH

  const int tid  = threadIdx.x;
  const int w    = tid >> 5;
  const int lane = tid & 31;
  const int half = lane >> 4;
  const int li   = lane & 15;
  const int tt   = blockIdx.x;       // t tile (16 rows)
  const int b    = blockIdx.y;

  v8f acc[4];
#pragma unroll
  for (int c = 0; c < 4; ++c) acc[c] = (v8f){0.f,0.f,0.f,0.f,0.f,0.f,0.f,0.f};

  for (int kb = 0; kb < 16; ++kb) {
    const int kOff = kb * 64;
    // A chunk: 16 rows x 64
    for (int idx = tid; idx < 256; idx += 128) {
      int i  = idx >> 4;
      int c4 = (idx & 15) << 2;
      int t  = tt * 16 + i; if (t > Tlen - 1) t = Tlen - 1;
      float4 v = *(const float4*)(A + ((size_t)b * Tlen + t) * Edim + kOff + c4);
      *(float4*)(As + i * WPITCH + c4) = v;
    }
    // gathered W chunk: Spad label rows x 64 (blank = row 0)
    for (int idx = tid; idx < Spad * 16; idx += 128) {
      int s  = idx >> 4;
      int c4 = (idx & 15) << 2;
      int lab = 0;
      if (s < Sext && (s & 1)) lab = ys[b * Llen + (s >> 1)];
      float4 v = *(const float4*)(W + (size_t)lab * Edim + kOff + c4);
      *(float4*)(Ws + s * WPITCH + c4) = v;
    }
    __syncthreads();
#pragma unroll
    for (int c = 0; c < 4; ++c) {
      int ct = w + 4 * c;            // wave-uniform -> EXEC stays full
      if (ct < 13) {
        const float* Ap = As + li * WPITCH + 2 * half;
        const float* Bp = Ws + (ct * 16 + li) * WPITCH + 2 * half;
#pragma unroll
        for (int kk = 0; kk < 16; ++kk) {
          v2f a  = *(const v2f*)(Ap + kk * 4);
          v2f bq = *(const v2f*)(Bp + kk * 4);
          acc[c] = __builtin_amdgcn_wmma_f32_16x16x4_f32(false, a, false, bq,
                                                         (short)0, acc[c],
                                                         false, false);
        }
      }
    }
    __syncthreads();
  }

#pragma unroll
  for (int c = 0; c < 4; ++c) {
    int ct = w + 4 * c;
    if (ct >= 13) continue;
    int scol = ct * 16 + li;
#pragma unroll
    for (int i = 0; i < 8; ++i) {
      int t = tt * 16 + (half ? i + 8 : i);
      if (t < Tlen && scol < Sext) {
        int lab = (scol & 1) ? ys[b * Llen + (scol >> 1)] : 0;
        float val = acc[c][i] + bias[lab] - rowlse[b * Tlen + t];
        lp[((size_t)b * Tlen + t) * Spad + scol] = val;
      }
    }
  }
}

// ---------------------------------------------------------------------------
// Kernel 3: CTC forward DP per batch. One block per b, alpha ping-pong in LDS.
// ---------------------------------------------------------------------------
__global__ void __launch_bounds__(256)
ctc_dp_kernel(const float* __restrict__ lp, const int* __restrict__ ys,
              float* __restrict__ ll) {
  __shared__ float al[2][Spad];
  const int b = blockIdx.x;
  const int s = threadIdx.x;
  const float* lpb = lp + (size_t)b * Tlen * Spad;

  bool can2 = false;
  if (s < Sext && (s & 1) && s >= 3) {
    int l1 = ys[b * Llen + ((s - 1) >> 1)];
    int l0 = ys[b * Llen + ((s - 3) >> 1)];
    can2 = (l1 != l0);
  }
  if (s < Spad) {
    al[0][s] = (s < 2) ? lpb[s] : NEGV;
    al[1][s] = NEGV;
  }
  __syncthreads();

  for (int t = 1; t < Tlen; ++t) {
    int p = (t - 1) & 1, q = t & 1;
    if (s < Sext) {
      float x  = al[p][s];
      float x1 = (s >= 1) ? al[p][s - 1] : NEGV;
      float x2 = can2 ? al[p][s - 2] : NEGV;
      float mx = fmaxf(fmaxf(x, x1), x2);
      float v  = mx + logf(expf(x - mx) + expf(x1 - mx) + expf(x2 - mx))
                 + lpb[(size_t)t * Spad + s];
      al[q][s] = v;
    }
    __syncthreads();
  }

  if (s == 0) {
    int q = (Tlen - 1) & 1;
    float x = al[q][Sext - 1], y = al[q][Sext - 2];
    float mx = fmaxf(x, y);
    ll[b] = mx + logf(expf(x - mx) + expf(y - mx));
  }
}

// ---------------------------------------------------------------------------
// Kernel 4: deterministic final reduction: out = -sum(ll)/B
// ---------------------------------------------------------------------------
__global__ void finalize_kernel(const float* __restrict__ ll, float* __restrict__ out) {
  if (threadIdx.x == 0 && blockIdx.x == 0) {
    float acc = 0.f;
    for (int i = 0; i < Bsz; ++i) acc += ll[i];
    out[0] = -acc / (float)Bsz;
  }
}

// ---------------------------------------------------------------------------
extern "C" void kernel_launch(void* const* d_in, const int* in_sizes, int n_in,
                              void* d_out, int out_size, void* d_ws, size_t ws_size,
                              hipStream_t stream) {
  (void)in_sizes; (void)n_in; (void)out_size; (void)ws_size;
  const float* hpad = (const float*)d_in[0];   // [B,T,E]
  const float* W    = (const float*)d_in[1];   // [V,E]
  const float* bias = (const float*)d_in[2];   // [V]
  const int*   ys   = (const int*)d_in[3];     // [B,L]
  float* out = (float*)d_out;

  float* ws     = (float*)d_ws;
  float* part   = ws;                                   // 16000*NSPLIT*2
  float* rowlse = ws + (size_t)Bsz * Tlen * NSPLIT * 2; // 16000
  float* lp     = rowlse + (size_t)Bsz * Tlen;          // 32*500*208
  float* ll     = lp + (size_t)Bsz * Tlen * Spad;       // 32

  const size_t lds1 = (64 * APITCH + 2 * 64 * WPITCH) * sizeof(float); // ~291 KB
  const size_t lds2 = (16 * WPITCH + Spad * WPITCH) * sizeof(float);   // ~60 KB

  gemm_rowstats_kernel<<<dim3(250, NSPLIT), 128, lds1, stream>>>(hpad, W, part);
  rowstats_merge_kernel<<<(Bsz * Tlen + 255) / 256, 256, 0, stream>>>(part, rowlse);
  gemm_labels_kernel<<<dim3(32, Bsz), 128, lds2, stream>>>(hpad, W, bias, ys, rowlse, lp);
  ctc_dp_kernel<<<Bsz, 256, 0, stream>>>(lp, ys, ll);
  finalize_kernel<<<1, 32, 0, stream>>>(ll, out);
}